// SolubilityGNN_80178449481894
// MI455X (gfx1250) — compile-verified
//
#include <hip/hip_runtime.h>

#define NNODES 100000
#define NEDGES 3200000
#define NG     4096
#define BN_EPS 1e-5f

typedef __attribute__((ext_vector_type(2))) float v2f;
typedef __attribute__((ext_vector_type(8))) float v8f;

// ---------------------------------------------------------------------------
// V_WMMA_F32_16X16X4_F32: D(16x16,f32) = A(16x4,f32) x B(4x16,f32) + C
// 8-arg pattern per probe: (neg_a, A, neg_b, B, c_mod, C, reuse_a, reuse_b)
// ---------------------------------------------------------------------------
__device__ __forceinline__ v8f wmma_f32_4(v2f a, v2f b, v8f c) {
  return __builtin_amdgcn_wmma_f32_16x16x4_f32(false, a, false, b, (short)0, c,
                                               false, false);
}

// ---------------------------------------------------------------------------
// Degree / dinv
// ---------------------------------------------------------------------------
__global__ void deg_kernel(const int* __restrict__ ei, float* __restrict__ deg) {
  int e = blockIdx.x * 256 + threadIdx.x;
  if (e < NEDGES) atomicAdd(&deg[ei[NEDGES + e]], 1.0f);
}

__global__ void dinv_kernel(float* __restrict__ deg) {
  int n = blockIdx.x * 256 + threadIdx.x;
  if (n < NNODES) deg[n] = rsqrtf(deg[n] + 1.0f);
}

// ---------------------------------------------------------------------------
// 64x64 GEMM via WMMA f32 16x16x4. One wave -> 16 rows x 64 cols.
// MODE 0: out = relu(in@W + bias)            (projection)
// MODE 1: out = in@W ; agg = dinv^2 * out    (conv xl + fused self-loop init)
// W is [64(k) x 64(c)] row-major; staged in LDS interleaved by k-pairs so a
// lane's B fragment {W[k][c], W[k+1][c]} is one ds_load_b64.
// ---------------------------------------------------------------------------
template <int MODE>
__global__ void gemm64_wmma(const float* __restrict__ in,
                            const float* __restrict__ W,
                            const float* __restrict__ bias,
                            const float* __restrict__ dinv,
                            float* __restrict__ out,
                            float* __restrict__ agg) {
  __shared__ float ldsW[64 * 64];
  const int tid = threadIdx.x;
#pragma unroll
  for (int i = 0; i < 16; ++i) {
    int idx = tid + i * 256;            // enumerates (k, c)
    int k = idx >> 6, c = idx & 63;
    ldsW[(k >> 1) * 128 + c * 2 + (k & 1)] = W[idx];
  }
  __syncthreads();

  const int wave = blockIdx.x * 8 + (tid >> 5);
  const int row0 = wave * 16;
  if (row0 >= NNODES) return;          // N = 100000 is a multiple of 16

  const int lane = tid & 31;
  const int ll = lane & 15;            // low lane id
  const int hi = lane >> 4;            // half-wave select
  const float* arow = in + (size_t)(row0 + ll) * 64 + 2 * hi;

  v8f acc0 = {}, acc1 = {}, acc2 = {}, acc3 = {};
#pragma unroll
  for (int ks = 0; ks < 16; ++ks) {    // K = 64 in steps of 4
    v2f a = *(const v2f*)(arow + 4 * ks);
    const float* bbase = &ldsW[(2 * ks + hi) * 128 + 2 * ll];
    acc0 = wmma_f32_4(a, *(const v2f*)(bbase +  0), acc0);
    acc1 = wmma_f32_4(a, *(const v2f*)(bbase + 32), acc1);
    acc2 = wmma_f32_4(a, *(const v2f*)(bbase + 64), acc2);
    acc3 = wmma_f32_4(a, *(const v2f*)(bbase + 96), acc3);
  }

  v8f accs[4] = {acc0, acc1, acc2, acc3};
#pragma unroll
  for (int t = 0; t < 4; ++t) {
    const int col = t * 16 + ll;
    const float bv = (MODE == 0) ? bias[col] : 0.0f;
#pragma unroll
    for (int j = 0; j < 8; ++j) {
      const int row = row0 + j + 8 * hi;   // C/D layout: M = j (+8 for hi half)
      float v = accs[t][j];
      if (MODE == 0) {
        out[(size_t)row * 64 + col] = fmaxf(v + bv, 0.0f);
      } else {
        out[(size_t)row * 64 + col] = v;
        const float di = dinv[row];
        agg[(size_t)row * 64 + col] = di * di * v;
      }
    }
  }
}

// ---------------------------------------------------------------------------
// Edge scatter: 16 threads per edge, float4 per thread.
// agg[dst] += xl[src] * dinv[src]*dinv[dst]     (L2-resident atomics)
// ---------------------------------------------------------------------------
__global__ void scatter_kernel(const int* __restrict__ ei,
                               const float* __restrict__ xl,
                               const float* __restrict__ dinv,
                               float* __restrict__ agg) {
  const int tid = blockIdx.x * 256 + threadIdx.x;   // NEDGES*16 threads
  const int e = tid >> 4;
  if (e >= NEDGES) return;
  const int c4 = (tid & 15) << 2;
  const int s = ei[e];
  const int d = ei[NEDGES + e];
  const float nrm = dinv[s] * dinv[d];
  const float4 v = *(const float4*)(xl + (size_t)s * 64 + c4);
  float* ap = agg + (size_t)d * 64 + c4;
  atomicAdd(ap + 0, v.x * nrm);
  atomicAdd(ap + 1, v.y * nrm);
  atomicAdd(ap + 2, v.z * nrm);
  atomicAdd(ap + 3, v.w * nrm);
}

// ---------------------------------------------------------------------------
// BN statistics over h_pre = agg + conv_b: per-channel sum / sumsq.
// Grid stride is a multiple of 64 -> each thread owns one fixed channel.
// ---------------------------------------------------------------------------
__global__ void bn_stats_kernel(const float* __restrict__ agg,
                                const float* __restrict__ bias,
                                float* __restrict__ stats) {
  __shared__ float s1[256];
  __shared__ float s2[256];
  const int tid = threadIdx.x;
  const int c = tid & 63;
  const float b = bias[c];
  float sum = 0.0f, sq = 0.0f;
  const int total = NNODES * 64;
  const int stride = gridDim.x * 256;   // multiple of 64
  for (int i = blockIdx.x * 256 + tid; i < total; i += stride) {
    const float v = agg[i] + b;
    sum += v;
    sq += v * v;
  }
  s1[tid] = sum;
  s2[tid] = sq;
  __syncthreads();
  if (tid < 64) {
    sum = s1[tid] + s1[tid + 64] + s1[tid + 128] + s1[tid + 192];
    sq  = s2[tid] + s2[tid + 64] + s2[tid + 128] + s2[tid + 192];
    atomicAdd(&stats[c], sum);
    atomicAdd(&stats[64 + c], sq);
  }
}

// ---------------------------------------------------------------------------
// h = relu((agg + b - mu) * rsqrt(var+eps) * g + bb) [+ residual AFTER relu]
// ---------------------------------------------------------------------------
__global__ void bn_apply_kernel(const float* __restrict__ agg,
                                const float* __restrict__ bias,
                                const float* __restrict__ g,
                                const float* __restrict__ bb,
                                const float* __restrict__ stats,
                                const float* __restrict__ residual,
                                float* __restrict__ out) {
  const int i = blockIdx.x * 256 + threadIdx.x;
  if (i >= NNODES * 64) return;
  const int c = i & 63;
  const float invN = 1.0f / (float)NNODES;
  const float mu = stats[c] * invN;
  const float var = stats[64 + c] * invN - mu * mu;
  float v = (agg[i] + bias[c] - mu) * rsqrtf(var + BN_EPS) * g[c] + bb[c];
  v = fmaxf(v, 0.0f);
  if (residual) v += residual[i];
  out[i] = v;
}

// ---------------------------------------------------------------------------
// Global mean pool (sums + counts via atomics into L2-hot 1MB buffer)
// ---------------------------------------------------------------------------
__global__ void pool_kernel(const float* __restrict__ h,
                            const int* __restrict__ batch,
                            float* __restrict__ pooled,
                            float* __restrict__ cnt) {
  const int tid = blockIdx.x * 256 + threadIdx.x;  // NNODES*16 threads
  const int n = tid >> 4;
  if (n >= NNODES) return;
  const int c4 = (tid & 15) << 2;
  const int b = batch[n];
  const float4 v = *(const float4*)(h + (size_t)n * 64 + c4);
  float* p = pooled + (size_t)b * 64 + c4;
  atomicAdd(p + 0, v.x);
  atomicAdd(p + 1, v.y);
  atomicAdd(p + 2, v.z);
  atomicAdd(p + 3, v.w);
  if (c4 == 0) atomicAdd(&cnt[b], 1.0f);
}

// ---------------------------------------------------------------------------
// Head MLP: out = relu(pooled@W1 + b1) @ W2 + b2   (64 -> 32 -> 1)
// ---------------------------------------------------------------------------
__global__ void head_kernel(const float* __restrict__ pooled,
                            const float* __restrict__ cnt,
                            const float* __restrict__ W1,
                            const float* __restrict__ b1,
                            const float* __restrict__ W2,
                            const float* __restrict__ b2,
                            float* __restrict__ out) {
  __shared__ float sW1[64 * 32];
  __shared__ float sb1[32];
  __shared__ float sW2[32];
  const int tid = threadIdx.x;
  for (int i = tid; i < 2048; i += 256) sW1[i] = W1[i];
  if (tid < 32) {
    sb1[tid] = b1[tid];
    sW2[tid] = W2[tid];
  }
  __syncthreads();
  const int gidx = blockIdx.x * 256 + tid;
  if (gidx >= NG) return;
  const float inv = 1.0f / fmaxf(cnt[gidx], 1.0f);
  float p[64];
#pragma unroll
  for (int k = 0; k < 64; ++k) p[k] = pooled[(size_t)gidx * 64 + k] * inv;
  float o = b2[0];
  for (int j = 0; j < 32; ++j) {
    float hv = sb1[j];
#pragma unroll
    for (int k = 0; k < 64; ++k) hv = fmaf(p[k], sW1[k * 32 + j], hv);
    o = fmaf(fmaxf(hv, 0.0f), sW2[j], o);
  }
  out[gidx] = o;
}

// ---------------------------------------------------------------------------
extern "C" void kernel_launch(void* const* d_in, const int* in_sizes, int n_in,
                              void* d_out, int out_size, void* d_ws,
                              size_t ws_size, hipStream_t stream) {
  (void)in_sizes; (void)n_in; (void)out_size; (void)ws_size;
  const float* x      = (const float*)d_in[0];
  const int*   ei     = (const int*)d_in[1];
  const int*   batch  = (const int*)d_in[2];
  const float* proj_W = (const float*)d_in[3];
  const float* proj_b = (const float*)d_in[4];
  const float* conv_W[3] = {(const float*)d_in[5], (const float*)d_in[7],
                            (const float*)d_in[9]};
  const float* conv_b[3] = {(const float*)d_in[6], (const float*)d_in[8],
                            (const float*)d_in[10]};
  const float* bn_g[3] = {(const float*)d_in[11], (const float*)d_in[13],
                          (const float*)d_in[15]};
  const float* bn_b[3] = {(const float*)d_in[12], (const float*)d_in[14],
                          (const float*)d_in[16]};
  const float* hW1 = (const float*)d_in[17];
  const float* hb1 = (const float*)d_in[18];
  const float* hW2 = (const float*)d_in[19];
  const float* hb2 = (const float*)d_in[20];

  float* ws     = (float*)d_ws;
  float* x0     = ws;                              // N*64
  float* hbuf   = x0 + (size_t)NNODES * 64;        // N*64
  float* xl     = hbuf + (size_t)NNODES * 64;      // N*64
  float* agg    = xl + (size_t)NNODES * 64;        // N*64
  float* dinv   = agg + (size_t)NNODES * 64;       // N (deg -> dinv in place)
  float* stats  = dinv + NNODES;                   // 128
  float* pooled = stats + 128;                     // G*64
  float* cnt    = pooled + (size_t)NG * 64;        // G

  // degree / dinv
  hipMemsetAsync(dinv, 0, NNODES * sizeof(float), stream);
  deg_kernel<<<(NEDGES + 255) / 256, 256, 0, stream>>>(ei, dinv);
  dinv_kernel<<<(NNODES + 255) / 256, 256, 0, stream>>>(dinv);

  const int gemm_blocks = (NNODES / 16 + 7) / 8;   // 782

  // x0 = relu(x @ proj_W + proj_b)
  gemm64_wmma<0><<<gemm_blocks, 256, 0, stream>>>(x, proj_W, proj_b, nullptr,
                                                  x0, nullptr);

  const float* cur = x0;
  for (int L = 0; L < 3; ++L) {
    // xl = cur @ W ; agg = dinv^2 * xl (self-loop fused)
    gemm64_wmma<1><<<gemm_blocks, 256, 0, stream>>>(cur, conv_W[L], nullptr,
                                                    dinv, xl, agg);
    // agg[dst] += xl[src] * dinv[src]*dinv[dst]
    scatter_kernel<<<(NEDGES * 16) / 256, 256, 0, stream>>>(ei, xl, dinv, agg);
    // BN stats over agg + conv_b
    hipMemsetAsync(stats, 0, 128 * sizeof(float), stream);
    bn_stats_kernel<<<512, 256, 0, stream>>>(agg, conv_b[L], stats);
    // normalize + relu (+ residual x0 after relu for layer 2)
    const float* res = (L == 1) ? x0 : nullptr;
    bn_apply_kernel<<<(NNODES * 64 + 255) / 256, 256, 0, stream>>>(
        agg, conv_b[L], bn_g[L], bn_b[L], stats, res, hbuf);
    cur = hbuf;
  }

  // global mean pool + head
  hipMemsetAsync(pooled, 0, (size_t)(NG * 64 + NG) * sizeof(float), stream);
  pool_kernel<<<(NNODES * 16 + 255) / 256, 256, 0, stream>>>(hbuf, batch,
                                                             pooled, cnt);
  head_kernel<<<(NG + 255) / 256, 256, 0, stream>>>(pooled, cnt, hW1, hb1,
                                                    hW2, hb2, (float*)d_out);
}